// CRFSeqLabeling_6305011990729
// MI455X (gfx1250) — compile-verified
//
#include <hip/hip_runtime.h>
#include <hip/hip_bf16.h>
#include <stddef.h>

// ---- problem constants (from reference) ----
// B=64 batch, S=512 seq, H=1024 hidden, L=9 labels
#define BN 64
#define SN 512
#define HN 1024
#define LN 9
#define LP 16            // padded label dim for WMMA / aligned stores
#define ROWS (BN * SN)   // 32768 flattened (b,s) rows
#define WSTR 1032        // LDS row stride (floats): 1032%64==8 -> b128 reads hit distinct banks

typedef __attribute__((ext_vector_type(2))) float v2f;
typedef __attribute__((ext_vector_type(8))) float v8f;

// -------------------------------------------------------------------------
// Kernel 1: logits[row][l] = sum_h inputs[row][h] * W[l][h] + bias[l]
// One wave32 computes a 16(row) x 16(label) tile with V_WMMA_F32_16X16X4_F32.
//
// K-slot assignment trick: across a PAIR of WMMAs, lanes 0-15 supply
// H {h..h+3} (K0,K1 of wmma#1 and wmma#2) and lanes 16-31 supply H
// {h+4..h+7} (K2,K3).  A and B use the same assignment, so the dot
// product over H is unchanged, but each lane now loads one contiguous
// float4 of A per 8 H-steps (global_load_b128, 512B/instr wave-wide).
//
// B (W) is staged in LDS once per block: rows 0..8 = W, row 9 = zeros
// (feeds the padded labels 9..15), so no select/multiply in the loop.
// -------------------------------------------------------------------------
__global__ void __launch_bounds__(256)
crf_logits_wmma_kernel(const float* __restrict__ inputs, // [ROWS][HN]
                       const float* __restrict__ Wm,     // [LN][HN]
                       const float* __restrict__ bias,   // [LN]
                       float* __restrict__ lg)           // [ROWS][LP]
{
    __shared__ float wlds[(LN + 1) * WSTR]; // ~41 KB of the 320 KB WGP LDS

    const int tid = threadIdx.x;

    // ---- cooperative LDS fill: 9 W rows + 1 zero row ----
    for (int i = tid; i < LN * HN; i += 256) {
        const int r = i >> 10;       // /HN
        const int c = i & (HN - 1);  // %HN
        wlds[r * WSTR + c] = Wm[i];
    }
    for (int i = tid; i < WSTR; i += 256) {
        wlds[LN * WSTR + i] = 0.0f;  // zero row for padded labels
    }
    __syncthreads();

    const int lane = tid & 31;
    const int wave = blockIdx.x * (blockDim.x >> 5) + (tid >> 5);
    const int rowBase = wave * 16;

    const int half = lane >> 4;   // 0: H {h..h+3}, 1: H {h+4..h+7}
    const int l    = lane & 15;   // A: row-in-tile   B: label column
    const int lw   = (l < LN) ? l : LN;  // padded labels -> zero row

    // A: row (rowBase+l), float4 index offset `half`
    const float4* ap =
        reinterpret_cast<const float4*>(inputs + (size_t)(rowBase + l) * HN) + half;
    // B: LDS row lw, float offset 4*half
    const float* bp = &wlds[lw * WSTR + 4 * half];

    v8f acc = {};
#pragma unroll 8
    for (int h = 0; h < HN; h += 8) {
        const float4 a4 = ap[h >> 2];                          // H h+4*half .. +3
        const float4 b4 = *reinterpret_cast<const float4*>(bp + h);
        v2f a0; a0[0] = a4.x; a0[1] = a4.y;   // K0,K1 (h0)   / K2,K3 (h1)
        v2f a1; a1[0] = a4.z; a1[1] = a4.w;   // K0,K1 of wmma#2
        v2f b0; b0[0] = b4.x; b0[1] = b4.y;
        v2f b1; b1[0] = b4.z; b1[1] = b4.w;
        // 8 args: (neg_a, A, neg_b, B, c_mod, C, reuse_a, reuse_b)
        acc = __builtin_amdgcn_wmma_f32_16x16x4_f32(
            false, a0, false, b0, (short)0, acc, false, false);
        acc = __builtin_amdgcn_wmma_f32_16x16x4_f32(
            false, a1, false, b1, (short)0, acc, false, false);
    }

    const float bias_l = (l < LN) ? bias[l] : 0.0f;
#pragma unroll
    for (int v = 0; v < 8; ++v) {
        const int row = rowBase + v + 8 * half; // D: VGPR v -> M=v / v+8
        lg[(size_t)row * LP + l] = acc[v] + bias_l;
    }
}

// -------------------------------------------------------------------------
// Kernel 2: per-batch CRF negative log-likelihood.
// One wave32 per batch element; lane j = label j (j<9 active).
// alpha recursion: new[j] = logsumexp_i(alpha[i] + T[i][j]) + emit[j]
// -------------------------------------------------------------------------
__global__ void __launch_bounds__(32)
crf_nll_kernel(const float* __restrict__ lg,      // [ROWS][LP]
               const int*   __restrict__ labels,  // [BN][SN]
               const int*   __restrict__ mask,    // [BN][SN]
               const float* __restrict__ trans,   // [LN][LN]
               const float* __restrict__ startS,  // [LN]
               const float* __restrict__ endS,    // [LN]
               float* __restrict__ nll)           // [BN]
{
    const int b = blockIdx.x;
    const int j = threadIdx.x;       // 0..31
    const bool act = (j < LN);
    const float NEG = -3.0e38f;

    // transition column j: tcol[i] = trans[i][j]
    float tcol[LN];
#pragma unroll
    for (int i = 0; i < LN; ++i) tcol[i] = act ? trans[i * LN + j] : 0.0f;

    const float* lgb = lg + (size_t)b * SN * LP;

    // ---- forward algorithm ----
    float alpha = act ? (lgb[j] + startS[j]) : NEG;
    for (int s = 1; s < SN; ++s) {
        const float emit = act ? lgb[(size_t)s * LP + j] : 0.0f;
        float v[LN];
#pragma unroll
        for (int i = 0; i < LN; ++i) v[i] = __shfl(alpha, i, 32) + tcol[i];
        float m = v[0];
#pragma unroll
        for (int i = 1; i < LN; ++i) m = fmaxf(m, v[i]);
        float sum = 0.0f;
#pragma unroll
        for (int i = 0; i < LN; ++i) sum += __expf(v[i] - m);
        const float na = m + __logf(sum) + emit;
        const int mt = mask[b * SN + s];
        if (act && mt > 0) alpha = na;
    }

    // norm = logsumexp_j(alpha + end[j])
    float w = act ? (alpha + endS[j]) : NEG;
    float m = w;
#pragma unroll
    for (int off = 16; off; off >>= 1) m = fmaxf(m, __shfl_xor(m, off, 32));
    float sum = __expf(w - m); // inactive lanes: exp(-huge) -> 0
#pragma unroll
    for (int off = 16; off; off >>= 1) sum += __shfl_xor(sum, off, 32);
    const float norm = m + __logf(sum);

    // ---- gold path score (parallel over s, lane-strided) ----
    float emitsum = 0.0f, transsum = 0.0f;
    int msum = 0;
    for (int s = j; s < SN; s += 32) {
        const int tag = labels[b * SN + s];
        const int mt  = mask[b * SN + s];
        if (mt > 0) {
            emitsum += lgb[(size_t)s * LP + tag];
            if (s >= 1) {
                const int tp = labels[b * SN + s - 1];
                transsum += trans[tp * LN + tag];
            }
            msum += 1;
        }
    }
#pragma unroll
    for (int off = 16; off; off >>= 1) {
        emitsum  += __shfl_xor(emitsum, off, 32);
        transsum += __shfl_xor(transsum, off, 32);
        msum     += __shfl_xor(msum, off, 32);
    }

    if (j == 0) {
        const int last = msum - 1;
        const int t0 = labels[b * SN];
        const int tl = labels[b * SN + last];
        const float score = emitsum + transsum + startS[t0] + endS[tl];
        nll[b] = norm - score;
    }
}

// -------------------------------------------------------------------------
// Kernel 3: mean over BN=64 batch NLLs -> scalar out[0]
// -------------------------------------------------------------------------
__global__ void __launch_bounds__(32)
crf_mean_kernel(const float* __restrict__ nll, float* __restrict__ out)
{
    const int j = threadIdx.x;
    float v = nll[j] + nll[j + 32];
#pragma unroll
    for (int off = 16; off; off >>= 1) v += __shfl_xor(v, off, 32);
    if (j == 0) out[0] = v * (1.0f / (float)BN);
}

// -------------------------------------------------------------------------
// Launcher.  Input order (setup_inputs dict order):
//   0 inputs [B,S,H] f32 | 1 labels [B,S] i32 | 2 mask [B,S] i32
//   3 W [L,H] f32 | 4 b [L] f32 | 5 trans_m [L,L] f32
//   6 start_scores [L] f32 | 7 end_scores [L] f32
// Output: 1 float (mean NLL).
// Workspace: logits [ROWS][16] f32 (2 MB) + nll [64] f32.
// -------------------------------------------------------------------------
extern "C" void kernel_launch(void* const* d_in, const int* in_sizes, int n_in,
                              void* d_out, int out_size, void* d_ws, size_t ws_size,
                              hipStream_t stream)
{
    (void)in_sizes; (void)n_in; (void)out_size; (void)ws_size;

    const float* inputs = (const float*)d_in[0];
    const int*   labels = (const int*)  d_in[1];
    const int*   mask   = (const int*)  d_in[2];
    const float* Wm     = (const float*)d_in[3];
    const float* bias   = (const float*)d_in[4];
    const float* trans  = (const float*)d_in[5];
    const float* startS = (const float*)d_in[6];
    const float* endS   = (const float*)d_in[7];

    float* lg  = (float*)d_ws;                       // ROWS*LP floats
    float* nll = lg + (size_t)ROWS * LP;             // BN floats
    float* out = (float*)d_out;

    // Phase 1: WMMA GEMM. 2048 waves of 16 rows each: 256 blocks x 256 thr.
    crf_logits_wmma_kernel<<<ROWS / (16 * 8), 256, 0, stream>>>(inputs, Wm, bias, lg);

    // Phase 2: one wave per batch element.
    crf_nll_kernel<<<BN, 32, 0, stream>>>(lg, labels, mask, trans, startS, endS, nll);

    // Phase 3: scalar mean.
    crf_mean_kernel<<<1, 32, 0, stream>>>(nll, out);
}